// GraphSAGE_8246337208621
// MI455X (gfx1250) — compile-verified
//
#include <hip/hip_runtime.h>
#include <hip/hip_bf16.h>

#define N_FEAT 128

typedef float v2f __attribute__((ext_vector_type(2)));
typedef float v8f __attribute__((ext_vector_type(8)));

// ---------------------------------------------------------------------------
// Scatter: msg[dst] += x[src] (mean-agg numerator), cnt[dst] += 1 (denominator)
// 32 lanes per edge; each lane handles 4 contiguous features (coalesced 512B
// row gather), then 4 global_atomic_add_f32 into the destination row. With the
// 51MB feature matrix L2-resident, these atomics are serviced by L2 atomic
// units at near-L2 rates.
// ---------------------------------------------------------------------------
__global__ void scatter_kernel(const float* __restrict__ x,
                               const int* __restrict__ src,
                               const int* __restrict__ dst,
                               float* __restrict__ msg,
                               float* __restrict__ cnt,
                               int nEdges, int addCnt) {
  int gid  = blockIdx.x * 256 + threadIdx.x;
  int e    = gid >> 5;
  if (e >= nEdges) return;
  int lane = gid & 31;
  int s = src[e];
  int d = dst[e];
  const float4 v = *(const float4*)(x + (size_t)s * N_FEAT + lane * 4);
  float* base = msg + (size_t)d * N_FEAT + lane * 4;
  atomicAdd(base + 0, v.x);
  atomicAdd(base + 1, v.y);
  atomicAdd(base + 2, v.z);
  atomicAdd(base + 3, v.w);
  if (addCnt && lane == 0) atomicAdd(cnt + d, 1.0f);
}

// ---------------------------------------------------------------------------
// Fused SAGE conv: y = act( (msg/max(cnt,1)) @ Wl + bias + xroot @ Wr )
// One block = 256 threads = 8 wave32s; block covers 16 node rows x 128 cols.
// Each wave owns one 16x16 tile and drives V_WMMA_F32_16X16X4_F32 over K=128.
//
// Operand striping per CDNA5 ISA 7.12.2 (h = lane>=16, l = lane&15):
//   A 16x4  : a.x = A[l][k0+2h],      a.y = A[l][k0+2h+1]
//   B 4x16  : b.x = B[k0+2h][n0+l],   b.y = B[k0+2h+1][n0+l]
//   C 16x16 : acc[r] = C[r + 8h][n0+l]
// ---------------------------------------------------------------------------
template <bool RELU>
__global__ __launch_bounds__(256)
void sage_gemm_kernel(const float* __restrict__ msg,    // [N,128] summed messages
                      const float* __restrict__ cnt,    // [N] degree (float)
                      const float* __restrict__ xroot,  // [N,128] root features
                      const float* __restrict__ Wl,     // [128,128] row-major (k,j)
                      const float* __restrict__ Wr,     // [128,128]
                      const float* __restrict__ bias,   // [128]
                      float* __restrict__ y,            // [N,128]
                      int nNodes) {
  __shared__ float sAgg[16][132];   // padded: stride 132 -> conflict-free reads
  __shared__ float sRoot[16][132];
  __shared__ float sRdeg[16];

  const int tid      = threadIdx.x;
  const int base     = blockIdx.x * 16;
  const bool fullTile = (base + 16 <= nNodes);   // uniform across block

  if (tid < 16) {
    int node = base + tid;
    float c = (fullTile || node < nNodes) ? cnt[node] : 1.0f;
    sRdeg[tid] = 1.0f / fmaxf(c, 1.0f);
  }
  __syncthreads();

  // Cooperative, coalesced stage of 16 rows of msg (scaled -> mean) and xroot.
  if (fullTile) {
    #pragma unroll
    for (int i = tid; i < 16 * N_FEAT; i += 256) {
      int r = i >> 7;
      int c = i & (N_FEAT - 1);
      size_t g = (size_t)(base + r) * N_FEAT + c;
      sAgg[r][c]  = msg[g] * sRdeg[r];
      sRoot[r][c] = xroot[g];
    }
  } else {
    for (int i = tid; i < 16 * N_FEAT; i += 256) {
      int r = i >> 7;
      int c = i & (N_FEAT - 1);
      int node = base + r;
      float a = 0.0f, b = 0.0f;
      if (node < nNodes) {
        size_t g = (size_t)node * N_FEAT + c;
        a = msg[g] * sRdeg[r];
        b = xroot[g];
      }
      sAgg[r][c]  = a;
      sRoot[r][c] = b;
    }
  }
  __syncthreads();

  const int wave = tid >> 5;
  const int lane = tid & 31;
  const int h    = lane >> 4;   // half-wave: selects K pair
  const int l    = lane & 15;
  const int n0   = wave * 16;   // 8 waves cover 128 output columns
  const int col  = n0 + l;

  v8f acc = {};

#if __has_builtin(__builtin_amdgcn_wmma_f32_16x16x4_f32)
  #pragma unroll 4
  for (int k0 = 0; k0 < N_FEAT; k0 += 4) {
    const int ka = k0 + 2 * h;
    v2f a, b;
    // aggregated-neighbor term: agg @ Wl
    a.x = sAgg[l][ka];
    a.y = sAgg[l][ka + 1];
    b.x = Wl[(size_t)ka * N_FEAT + col];
    b.y = Wl[(size_t)(ka + 1) * N_FEAT + col];
    acc = __builtin_amdgcn_wmma_f32_16x16x4_f32(false, a, false, b,
                                                (short)0, acc, false, false);
    // root term: xroot @ Wr
    a.x = sRoot[l][ka];
    a.y = sRoot[l][ka + 1];
    b.x = Wr[(size_t)ka * N_FEAT + col];
    b.y = Wr[(size_t)(ka + 1) * N_FEAT + col];
    acc = __builtin_amdgcn_wmma_f32_16x16x4_f32(false, a, false, b,
                                                (short)0, acc, false, false);
  }
#else
  // Scalar fallback (keeps kernel compiling if the f32 WMMA builtin is absent)
  #pragma unroll
  for (int r = 0; r < 8; ++r) {
    int m = r + h * 8;
    float s = 0.0f;
    for (int k = 0; k < N_FEAT; ++k)
      s = fmaf(sAgg[m][k],  Wl[(size_t)k * N_FEAT + col],
          fmaf(sRoot[m][k], Wr[(size_t)k * N_FEAT + col], s));
    acc[r] = s;
  }
#endif

  const float bv = bias[col];
  if (fullTile) {
    // Branch-free epilogue: one row pointer, 8 stores at immediate offsets
    // (row stride 512B fits the 24-bit IOFFSET; stores coalesce per half-wave).
    float* yp = y + (size_t)(base + h * 8) * N_FEAT + col;
    #pragma unroll
    for (int r = 0; r < 8; ++r) {
      float v = acc[r] + bv;
      if (RELU) v = fmaxf(v, 0.0f);
      yp[(size_t)r * N_FEAT] = v;
    }
  } else {
    #pragma unroll
    for (int r = 0; r < 8; ++r) {
      int node = base + r + h * 8;
      if (node < nNodes) {
        float v = acc[r] + bv;
        if (RELU) v = fmaxf(v, 0.0f);
        y[(size_t)node * N_FEAT + col] = v;
      }
    }
  }
}

// ---------------------------------------------------------------------------
// Link predictor: out[i] = x2[s]·Wlin[0:128] + x2[d]·Wlin[128:256] + blin
// One wave per label; strided coalesced loads + wave32 butterfly reduction.
// ---------------------------------------------------------------------------
__global__ void predict_kernel(const float* __restrict__ x,
                               const int* __restrict__ sIdx,
                               const int* __restrict__ dIdx,
                               const float* __restrict__ Wlin,
                               const float* __restrict__ blin,
                               float* __restrict__ out, int nLabels) {
  int gid  = blockIdx.x * 256 + threadIdx.x;
  int lbl  = gid >> 5;
  if (lbl >= nLabels) return;
  int lane = gid & 31;
  int s = sIdx[lbl];
  int d = dIdx[lbl];
  float acc = 0.0f;
  #pragma unroll
  for (int f = lane; f < N_FEAT; f += 32) {
    acc = fmaf(x[(size_t)s * N_FEAT + f], Wlin[f],
          fmaf(x[(size_t)d * N_FEAT + f], Wlin[N_FEAT + f], acc));
  }
  #pragma unroll
  for (int off = 16; off; off >>= 1)
    acc += __shfl_xor(acc, off, 32);
  if (lane == 0) out[lbl] = acc + blin[0];
}

// ---------------------------------------------------------------------------
extern "C" void kernel_launch(void* const* d_in, const int* in_sizes, int n_in,
                              void* d_out, int out_size, void* d_ws, size_t ws_size,
                              hipStream_t stream) {
  const int*   ei   = (const int*)d_in[0];    // edge_index [2,E]
  const int*   eli  = (const int*)d_in[1];    // edge_label_index [2,L]
  const float* emb  = (const float*)d_in[2];  // [N,128]
  const float* W1l  = (const float*)d_in[3];
  const float* b1   = (const float*)d_in[4];
  const float* W1r  = (const float*)d_in[5];
  const float* W2l  = (const float*)d_in[6];
  const float* b2   = (const float*)d_in[7];
  const float* W2r  = (const float*)d_in[8];
  const float* Wlin = (const float*)d_in[9];  // [256]
  const float* blin = (const float*)d_in[10]; // [1]

  const int E = in_sizes[0] / 2;
  const int L = in_sizes[1] / 2;
  const int N = in_sizes[2] / N_FEAT;

  const int* src = ei;
  const int* dst = ei + E;
  const int* ls  = eli;
  const int* ld  = eli + L;

  // Workspace layout: msg[N*128] | x1[N*128] | x2[N*128] | cnt[N]
  float* msg = (float*)d_ws;
  float* x1  = msg + (size_t)N * N_FEAT;
  float* x2  = x1  + (size_t)N * N_FEAT;
  float* cnt = x2  + (size_t)N * N_FEAT;

  const int scatterBlocks = (E * 32 + 255) / 256;
  const int convBlocks    = (N + 15) / 16;
  const int predBlocks    = (L * 32 + 255) / 256;

  // ---- layer 1 ----
  hipMemsetAsync(msg, 0, (size_t)N * N_FEAT * sizeof(float), stream);
  hipMemsetAsync(cnt, 0, (size_t)N * sizeof(float), stream);
  scatter_kernel<<<scatterBlocks, 256, 0, stream>>>(emb, src, dst, msg, cnt, E, 1);
  sage_gemm_kernel<true><<<convBlocks, 256, 0, stream>>>(msg, cnt, emb, W1l, W1r,
                                                         b1, x1, N);
  // ---- layer 2 (reuse degree counts) ----
  hipMemsetAsync(msg, 0, (size_t)N * N_FEAT * sizeof(float), stream);
  scatter_kernel<<<scatterBlocks, 256, 0, stream>>>(x1, src, dst, msg, cnt, E, 0);
  sage_gemm_kernel<false><<<convBlocks, 256, 0, stream>>>(msg, cnt, x1, W2l, W2r,
                                                          b2, x2, N);
  // ---- link predictor ----
  predict_kernel<<<predBlocks, 256, 0, stream>>>(x2, ls, ld, Wlin, blin,
                                                 (float*)d_out, L);
}